// LineOutlinerTsa_68917045232104
// MI455X (gfx1250) — compile-verified
//
#include <hip/hip_runtime.h>
#include <hip/hip_bf16.h>
#include <math.h>

// ---------------------------------------------------------------------------
// LineOutlinerTsa on MI455X (gfx1250, wave32).
// fp32 end-to-end; conv layers = implicit GEMM on V_WMMA_F32_16X16X4_F32.
// All conv geometry is compile-time (templates): no runtime integer division,
// power-of-two Wo -> shifts, dual accumulators to hide WMMA->WMMA RAW latency.
// ---------------------------------------------------------------------------

typedef __attribute__((ext_vector_type(2))) float v2f;
typedef __attribute__((ext_vector_type(8))) float v8f;

#define IMG_H 2400
#define IMG_W 3600
#define PS    64
#define PI_F  3.14159265358979323846f

// ---------------------------------------------------------------------------
// Kernel 0: softmax(tsa_w) + rotated bilinear patch extraction + temporal blend
// Writes blended input image (3, 64, 64).
// ---------------------------------------------------------------------------
__global__ void k_blend(const float* __restrict__ img,
                        const float* __restrict__ steps,
                        const float* __restrict__ tsa_w,
                        const int*   __restrict__ sol_index,
                        float* __restrict__ out)
{
    int idx = blockIdx.x * blockDim.x + threadIdx.x;
    if (idx >= 3 * PS * PS) return;
    int c  = idx / (PS * PS);
    int p  = idx % (PS * PS);
    int py = p / PS, px = p % PS;

    float w0 = tsa_w[0], w1 = tsa_w[1], w2 = tsa_w[2];
    float mx = fmaxf(w0, fmaxf(w1, w2));
    float e0 = expf(w0 - mx), e1 = expf(w1 - mx), e2 = expf(w2 - mx);
    float inv = 1.0f / (e0 + e1 + e2);
    float att[3] = { e0 * inv, e1 * inv, e2 * inv };

    int si    = sol_index[0];
    int start = si - 3; if (start < 0) start = 0;
    int cnt   = si - start + 1;

    const float* ip = img + (size_t)c * IMG_H * IMG_W;

    float acc = 0.0f;
    for (int t = 0; t < 3; ++t) {
        int fr = start + (cnt - 3) + t;       // frames = patches[-3:]
        if (fr < start) fr = start;
        const float* s = steps + fr * 8;
        float ux = s[0], uy = s[1];           // upper
        float cx = s[2], cy = s[3];           // base (patch center)
        float ang = s[6];
        float ddx = ux - cx, ddy = uy - cy;
        float h = fmaxf(64.0f, sqrtf(ddx * ddx + ddy * ddy));
        float th = -ang * PI_F / 180.0f;
        float ct = cosf(th), st = sinf(th);
        float stepsz = h * 5.0f / 64.0f;
        float dx = ((float)px - 31.5f) * stepsz;
        float dy = ((float)py - 31.5f) * stepsz;
        float X = cx + dx * ct - dy * st;
        float Y = cy + dx * st + dy * ct;
        float x0 = floorf(X), y0 = floorf(Y);
        float wx = X - x0,   wy = Y - y0;
        int x0i = (int)x0, y0i = (int)y0;
        int x1i = x0i + 1,  y1i = y0i + 1;
        x0i = min(max(x0i, 0), IMG_W - 1);
        x1i = min(max(x1i, 0), IMG_W - 1);
        y0i = min(max(y0i, 0), IMG_H - 1);
        y1i = min(max(y1i, 0), IMG_H - 1);
        float v00 = ip[(size_t)y0i * IMG_W + x0i];
        float v01 = ip[(size_t)y0i * IMG_W + x1i];
        float v10 = ip[(size_t)y1i * IMG_W + x0i];
        float v11 = ip[(size_t)y1i * IMG_W + x1i];
        float v = (1.0f - wy) * ((1.0f - wx) * v00 + wx * v01)
                +         wy  * ((1.0f - wx) * v10 + wx * v11);
        acc += att[t] * v;
    }
    out[idx] = acc;
}

// ---------------------------------------------------------------------------
// 3x3 conv (pad=1) as implicit GEMM on V_WMMA_F32_16X16X4_F32.
//   M = Cout, N = Ho*Wo, K = Cin*9; one wave = one 16x16 C tile.
// Fragment layouts per CDNA5 ISA 7.12.2 (32-bit, wave32):
//   A: lane l -> row l&15; K pair {0,1} (lanes 0-15) / {2,3} (lanes 16-31).
//   B: lane l -> col l&15; same K pairing.
//   C/D: VGPR r -> row r + (l>>4)*8, col l&15.
// ---------------------------------------------------------------------------
template<int CIN, int H, int W, int COUT, int HO, int WO, int STRIDE, bool RELU>
__global__ __launch_bounds__(32)
void k_conv_wmma(const float* __restrict__ in, const float* __restrict__ w,
                 float* __restrict__ out)
{
    constexpr int K   = CIN * 9;
    constexpr int N   = HO * WO;
    constexpr int NTN = (N + 15) / 16;

    const int tm  = (int)blockIdx.x / NTN;   // compile-time NTN -> cheap
    const int tn  = (int)blockIdx.x % NTN;
    const int oc0 = tm << 4;
    const int n0  = tn << 4;

    const int lane = threadIdx.x;            // 0..31
    const int r16  = lane & 15;
    const int hi   = lane >> 4;              // K pair select

    // B column for this lane
    const int  n      = n0 + r16;
    const bool nvalid = (N % 16 == 0) ? true : (n < N);
    const int  nn     = nvalid ? n : 0;
    const int  oy     = nn / WO;             // WO is power of two -> shift
    const int  ox     = nn - oy * WO;
    const int  iy0    = oy * STRIDE - 1;     // pad = 1
    const int  ix0    = ox * STRIDE - 1;

    // A row for this lane
    const float* wrow = w + (size_t)(oc0 + r16) * K;

    // im2col gather of B(k, n): k -> (ic, ky, kx); zero outside image/pad.
    auto bval = [&](int k) -> float {
        if (N % 16 != 0 && !nvalid) return 0.0f;
        int ic = k / 9;                      // constant divisor -> mul/shift
        int r9 = k - ic * 9;
        int ky = r9 / 3;
        int kx = r9 - ky * 3;
        int iy = iy0 + ky, ix = ix0 + kx;
        if (iy < 0 || iy >= H || ix < 0 || ix >= W) return 0.0f;
        return in[(ic * H + iy) * W + ix];
    };

    v8f acc0 = { 0.f, 0.f, 0.f, 0.f, 0.f, 0.f, 0.f, 0.f };

    if constexpr ((K % 8) == 0) {
        // Fast path: no K guards, aligned float2 weight loads, dual
        // accumulators to break the WMMA->WMMA RAW dependency chain.
        v8f acc1 = { 0.f, 0.f, 0.f, 0.f, 0.f, 0.f, 0.f, 0.f };
#pragma unroll 2
        for (int k0 = 0; k0 < K; k0 += 8) {
            __builtin_prefetch(wrow + k0 + 128, 0, 0);  // global_prefetch_b8
            {
                const int kk = k0 + (hi << 1);
                v2f a = *(const v2f*)(wrow + kk);
                v2f b = { bval(kk), bval(kk + 1) };
                acc0 = __builtin_amdgcn_wmma_f32_16x16x4_f32(
                           false, a, false, b, (short)0, acc0, false, false);
            }
            {
                const int kk = k0 + 4 + (hi << 1);
                v2f a = *(const v2f*)(wrow + kk);
                v2f b = { bval(kk), bval(kk + 1) };
                acc1 = __builtin_amdgcn_wmma_f32_16x16x4_f32(
                           false, a, false, b, (short)0, acc1, false, false);
            }
        }
        acc0 = acc0 + acc1;
    } else {
        // Guarded path (conv1: K = 27).
        for (int k0 = 0; k0 < K; k0 += 4) {
            const int kk = k0 + (hi << 1);
            float ax = (kk     < K) ? wrow[kk]     : 0.0f;
            float ay = (kk + 1 < K) ? wrow[kk + 1] : 0.0f;
            float bx = (kk     < K) ? bval(kk)     : 0.0f;
            float by = (kk + 1 < K) ? bval(kk + 1) : 0.0f;
            v2f a = { ax, ay };
            v2f b = { bx, by };
            acc0 = __builtin_amdgcn_wmma_f32_16x16x4_f32(
                       false, a, false, b, (short)0, acc0, false, false);
        }
    }

#pragma unroll
    for (int r = 0; r < 8; ++r) {
        const int row = r + (hi << 3);
        if (N % 16 == 0 || n < N) {
            float v = acc0[r];
            if (RELU) v = fmaxf(v, 0.0f);
            out[(size_t)(oc0 + row) * N + n] = v;
        }
    }
}

// ---------------------------------------------------------------------------
// LSTM gate block: g = (512, 8, 8) -> i/f/g/o (128 each);
// c = sigmoid(i)*tanh(g); y = sigmoid(o)*tanh(c).   (f unused: no prior cell)
// ---------------------------------------------------------------------------
__global__ void k_lstm(const float* __restrict__ g, float* __restrict__ y)
{
    int idx = blockIdx.x * blockDim.x + threadIdx.x;
    if (idx >= 128 * 64) return;
    int c = idx >> 6, n = idx & 63;
    float ig = g[(size_t)c * 64 + n];
    float gg = g[(size_t)(c + 256) * 64 + n];
    float og = g[(size_t)(c + 384) * 64 + n];
    float cc = (1.0f / (1.0f + expf(-ig))) * tanhf(gg);
    y[idx]   = (1.0f / (1.0f + expf(-og))) * tanhf(cc);
}

// ---------------------------------------------------------------------------
// Head: global-avg-pool (512,2,2) -> fc (8x512) -> sigmoid[7] -> geometry.
// ---------------------------------------------------------------------------
__global__ void k_head(const float* __restrict__ y5,
                       const float* __restrict__ fc_w,
                       const float* __restrict__ fc_b,
                       const float* __restrict__ steps,
                       const int*   __restrict__ sol_index,
                       float* __restrict__ outp)
{
    __shared__ float feat[512];
    __shared__ float o8[8];
    int t = threadIdx.x;
    if (t < 512) {
        const float* p = y5 + (size_t)t * 4;
        feat[t] = 0.25f * (p[0] + p[1] + p[2] + p[3]);
    }
    __syncthreads();
    if (t < 8) {
        float s = fc_b[t];
        const float* wr = fc_w + (size_t)t * 512;
        for (int c = 0; c < 512; ++c) s += wr[c] * feat[c];
        o8[t] = s;
    }
    __syncthreads();
    if (t == 0) {
        int si = sol_index[0];
        const float* s = steps + si * 8;      // su[-1]
        float ux = s[0], uy = s[1];
        float bx = s[2], by = s[3];
        float ang0 = s[6];
        float ddx = ux - bx, ddy = uy - by;
        float h  = fmaxf(64.0f, sqrtf(ddx * ddx + ddy * ddy));
        float th = ang0 * PI_F / 180.0f;
        float ct = cosf(th), st = sinf(th);
        float scale = 5.0f * h / 64.0f;
        // v @ R.T @ (scale*I)  ==  scale * (R @ v)
        float bpx = scale * (ct * o8[0] - st * o8[1]);
        float bpy = scale * (st * o8[0] + ct * o8[1]);
        float upx = scale * (ct * o8[3] - st * o8[4]) + bx;
        float upy = scale * (st * o8[3] + ct * o8[4]) + by;
        float lpx = scale * (ct * o8[5] - st * o8[6]) + bx;
        float lpy = scale * (st * o8[5] + ct * o8[6]) + by;
        outp[0] = upx;        outp[1] = upy;            // up
        outp[2] = bx + bpx;   outp[3] = by + bpy;       // nb
        outp[4] = lpx;        outp[5] = lpy;            // lp
        outp[6] = ang0 + o8[2]; outp[7] = 0.0f;         // [na, 0]
        outp[8] = 1.0f / (1.0f + expf(-o8[7]));         // sigmoid(out8[7])
        outp[9] = 0.0f;
    }
}

// ---------------------------------------------------------------------------
// Launch: 9 kernels, ping-pong two 128KB slabs in d_ws (256KB total; all
// intermediates stay resident in MI455X's 192MB L2).
// ---------------------------------------------------------------------------
template<int COUT, int HO, int WO>
constexpr int conv_grid() { return (COUT / 16) * ((HO * WO + 15) / 16); }

extern "C" void kernel_launch(void* const* d_in, const int* in_sizes, int n_in,
                              void* d_out, int out_size, void* d_ws, size_t ws_size,
                              hipStream_t stream)
{
    const float* img   = (const float*)d_in[0];
    const float* steps = (const float*)d_in[1];
    const float* tsa_w = (const float*)d_in[2];
    const float* w1    = (const float*)d_in[3];
    const float* w2    = (const float*)d_in[4];
    const float* w3    = (const float*)d_in[5];
    const float* wl    = (const float*)d_in[6];
    const float* w4    = (const float*)d_in[7];
    const float* w5    = (const float*)d_in[8];
    const float* fc_w  = (const float*)d_in[9];
    const float* fc_b  = (const float*)d_in[10];
    const int*   si    = (const int*)d_in[11];
    float* out = (float*)d_out;

    float* A = (float*)d_ws;     // 32768 floats
    float* B = A + 32768;        // 32768 floats

    // blend (3,64,64) -> A
    k_blend<<<48, 256, 0, stream>>>(img, steps, tsa_w, si, A);
    // conv1: (3,64,64) -> (32,32,32)
    k_conv_wmma<3, 64, 64, 32, 32, 32, 2, true>
        <<<conv_grid<32, 32, 32>(), 32, 0, stream>>>(A, w1, B);
    // conv2: (32,32,32) -> (64,16,16)
    k_conv_wmma<32, 32, 32, 64, 16, 16, 2, true>
        <<<conv_grid<64, 16, 16>(), 32, 0, stream>>>(B, w2, A);
    // conv3: (64,16,16) -> (128,8,8)
    k_conv_wmma<64, 16, 16, 128, 8, 8, 2, true>
        <<<conv_grid<128, 8, 8>(), 32, 0, stream>>>(A, w3, B);
    // lstm conv: (128,8,8) -> (512,8,8), stride 1, linear
    k_conv_wmma<128, 8, 8, 512, 8, 8, 1, false>
        <<<conv_grid<512, 8, 8>(), 32, 0, stream>>>(B, wl, A);
    // gates: g(A) -> y(B)
    k_lstm<<<(128 * 64) / 256, 256, 0, stream>>>(A, B);
    // conv4: (128,8,8) -> (256,4,4)
    k_conv_wmma<128, 8, 8, 256, 4, 4, 2, true>
        <<<conv_grid<256, 4, 4>(), 32, 0, stream>>>(B, w4, A);
    // conv5: (256,4,4) -> (512,2,2)
    k_conv_wmma<256, 4, 4, 512, 2, 2, 2, true>
        <<<conv_grid<512, 2, 2>(), 32, 0, stream>>>(A, w5, B);
    // head
    k_head<<<1, 512, 0, stream>>>(B, fc_w, fc_b, steps, si, out);
}